// ROI_Pooling_5823975653464
// MI455X (gfx1250) — compile-verified
//
#include <hip/hip_runtime.h>
#include <float.h>

// ROI max-pool, fixed problem: B=8, C=128, H=W=50, crop 7x7, R = rois count.
// Grid (R, C/4): each block = one ROI x 4 channels. Single async global->LDS
// stage (gfx1250 async data-mover), one wait+barrier, then 196 threads each
// produce one (channel, bin) max. No serial chunk loop -> critical path is a
// single memory round-trip; 8192 blocks saturate the device.

#define CROP      7
#define CH_CHUNK  4
#define NTHREADS  256

__global__ __launch_bounds__(NTHREADS) void roi_pool_gfx1250(
    const float* __restrict__ input,   // [B=8][C=128][H=50][W=50]
    const int*   __restrict__ rois,    // [R][5] = b, y1, x1, y2, x2
    float*       __restrict__ out)     // [R][128][7][7]
{
    constexpr int C = 128, H = 50, W = 50, HW = H * W;

    __shared__ float smem[CH_CHUNK * HW];   // 40 KB: 4 channels x 50x50 band window

    const int r   = blockIdx.x;
    const int c0  = blockIdx.y * CH_CHUNK;
    const int tid = threadIdx.x;

    const int b  = rois[r * 5 + 0];
    const int y1 = rois[r * 5 + 1];
    const int x1 = rois[r * 5 + 2];
    const int y2 = rois[r * 5 + 3];
    const int x2 = rois[r * 5 + 4];
    const int rh = y2 - y1 + 1;             // >= 1 (coords pre-sorted)
    const int rw = x2 - x1 + 1;             // >= 1

    // LDS byte address of smem base (generic shared ptr: low 32 bits = LDS offset)
    const unsigned smemBase = (unsigned)(unsigned long long)(&smem[0]);

    // 8-byte packets per channel band: rh rows * 50 floats (always even count)
    const int n64 = rh * (W / 2);

    // ---- stage 4 channel row-bands [y1..y2] x [0..49] into LDS, async ----
    for (int ch = 0; ch < CH_CHUNK; ++ch) {
        const float* gsrc = input + (size_t)(b * C + c0 + ch) * HW + y1 * W;
        const unsigned ldsCh = smemBase + (unsigned)(ch * HW * 4);
        for (int i = tid; i < n64; i += NTHREADS) {
            const unsigned lds = ldsCh + (unsigned)(i * 8);
            const float*   g   = gsrc + i * 2;
            asm volatile("global_load_async_to_lds_b64 %0, %1, off"
                         :: "v"(lds), "v"(g) : "memory");
        }
    }
    asm volatile("s_wait_asynccnt 0" ::: "memory");  // this wave's copies complete
    __syncthreads();                                 // all waves' copies visible

    // ---- 196 threads: one (channel, bin) each ----
    if (tid < CH_CHUNK * CROP * CROP) {
        const int ch  = tid / (CROP * CROP);
        const int bin = tid - ch * (CROP * CROP);
        const int bi  = bin / CROP;
        const int bj  = bin - bi * CROP;

        // bin bounds in band-local coords, matching reference floor/ceil math:
        // s = floor(i*len/7), e = ceil((i+1)*len/7)
        const int ys = (bi * rh) / CROP;
        const int ye = ((bi + 1) * rh + (CROP - 1)) / CROP;
        const int xs = (bj * rw) / CROP;
        const int xe = ((bj + 1) * rw + (CROP - 1)) / CROP;

        const float* sp = &smem[ch * HW];
        float m = -FLT_MAX;
        for (int y = ys; y < ye; ++y) {
            const float* row = sp + y * W + x1;
            for (int x = xs; x < xe; ++x)
                m = fmaxf(m, row[x]);
        }
        out[((size_t)r * C + (c0 + ch)) * (CROP * CROP) + bin] = m;
    }
}

extern "C" void kernel_launch(void* const* d_in, const int* in_sizes, int n_in,
                              void* d_out, int out_size, void* d_ws, size_t ws_size,
                              hipStream_t stream) {
    const float* input = (const float*)d_in[0];
    const int*   rois  = (const int*)d_in[1];
    float*       out   = (float*)d_out;

    const int R = in_sizes[1] / 5;   // 256 ROIs
    constexpr int C = 128;
    (void)n_in; (void)out_size; (void)d_ws; (void)ws_size;

    roi_pool_gfx1250<<<dim3(R, C / CH_CHUNK), dim3(NTHREADS), 0, stream>>>(input, rois, out);
}